// LSTMCell_46059229282552
// MI455X (gfx1250) — compile-verified
//
#include <hip/hip_runtime.h>
#include <math.h>

// ---------------------------------------------------------------------------
// Batch-normalized LSTM cell for MI455X (gfx1250, wave32).
//   B = 32768, H = I = 256, 4H = 1024.
// GEMMs done with V_WMMA_F32_16X16X4_F32 (fp32 matrix path: problem is
// memory-bound at 23.3 TB/s, so fp32 keeps reference precision for free).
// BN statistics are computed with deterministic two-level reductions
// (no float atomics) so every call gives bit-identical output.
// ---------------------------------------------------------------------------

typedef float v2f __attribute__((ext_vector_type(2)));
typedef float v8f __attribute__((ext_vector_type(8)));

#define BROWS 32768
#define HID   256
#define KIN   256
#define G4    1024        // 4*H
#define EPSV  1e-5f
#define NCHUNK 64         // row chunks for deterministic stats (512 rows each)

// ---------------------------------------------------------------------------
// Kernel 1: dual GEMM.  D[b,n] = sum_k A[b,k] * W[n,k]   (W row-major [4H,K])
// grid = (B/128, 4H/64, 2);  block = 256 threads (8 waves).
// Each wave: 16 rows x 64 cols (4 WMMA accumulators), K-loop step 4.
//
// WMMA f32 16x16x4 operand layout (ISA 7.12.2):
//   A (16x4):  lanes 0-15 -> M=lane,    V0/V1 = K=k,k+1
//              lanes 16-31-> M=lane-16, V0/V1 = K=k+2,k+3
//   B (4x16):  mirrored (lane = N, half selects K pair)
// Both are satisfied by one float2 load at  row*K + k + (lane/16)*2.
// ---------------------------------------------------------------------------
__global__ __launch_bounds__(256)
void gemm_gates(const float* __restrict__ X, const float* __restrict__ Hst,
                const float* __restrict__ Wih, const float* __restrict__ Whh,
                float* __restrict__ ih, float* __restrict__ hh)
{
    const int wave = threadIdx.x >> 5;
    const int lane = threadIdx.x & 31;
    const int half = lane >> 4;      // 0: lanes 0-15, 1: lanes 16-31
    const int l16  = lane & 15;

    const float* __restrict__ A = blockIdx.z ? Hst : X;
    const float* __restrict__ W = blockIdx.z ? Whh : Wih;
    float* __restrict__ D       = blockIdx.z ? hh  : ih;

    const int m0 = (blockIdx.x * 8 + wave) * 16;
    const int n0 = blockIdx.y * 64;

    v8f acc0 = {}, acc1 = {}, acc2 = {}, acc3 = {};

    const float* ap  = A + (size_t)(m0 + l16) * KIN + half * 2;
    const float* bp0 = W + (size_t)(n0 +  0 + l16) * KIN + half * 2;
    const float* bp1 = W + (size_t)(n0 + 16 + l16) * KIN + half * 2;
    const float* bp2 = W + (size_t)(n0 + 32 + l16) * KIN + half * 2;
    const float* bp3 = W + (size_t)(n0 + 48 + l16) * KIN + half * 2;

#pragma unroll 4
    for (int k = 0; k < KIN; k += 4) {
        v2f a  = *(const v2f*)(ap  + k);
        v2f b0 = *(const v2f*)(bp0 + k);
        v2f b1 = *(const v2f*)(bp1 + k);
        v2f b2 = *(const v2f*)(bp2 + k);
        v2f b3 = *(const v2f*)(bp3 + k);
        acc0 = __builtin_amdgcn_wmma_f32_16x16x4_f32(false, a, false, b0,
                                                     (short)0, acc0, false, false);
        acc1 = __builtin_amdgcn_wmma_f32_16x16x4_f32(false, a, false, b1,
                                                     (short)0, acc1, false, false);
        acc2 = __builtin_amdgcn_wmma_f32_16x16x4_f32(false, a, false, b2,
                                                     (short)0, acc2, false, false);
        acc3 = __builtin_amdgcn_wmma_f32_16x16x4_f32(false, a, false, b3,
                                                     (short)0, acc3, false, false);
    }

    // C/D layout: VGPR r, lanes 0-15 -> M=m0+r, lanes 16-31 -> M=m0+8+r; N=l16.
#pragma unroll
    for (int r = 0; r < 8; ++r) {
        size_t base = (size_t)(m0 + half * 8 + r) * G4 + n0 + l16;
        D[base +  0] = acc0[r];
        D[base + 16] = acc1[r];
        D[base + 32] = acc2[r];
        D[base + 48] = acc3[r];
    }
}

// ---------------------------------------------------------------------------
// Kernel 2: per-column partial sums / sumsq over row chunks (deterministic).
// grid = (4, NCHUNK, 2); block = 256.  part layout: [a][sum|sq][chunk][1024]
// ---------------------------------------------------------------------------
__global__ __launch_bounds__(256)
void gate_stats_partial(const float* __restrict__ ih, const float* __restrict__ hh,
                        float* __restrict__ part)
{
    const float* src = blockIdx.z ? hh : ih;
    const int col   = blockIdx.x * 256 + threadIdx.x;   // 0..1023
    const int chunk = blockIdx.y;
    const int rows  = BROWS / NCHUNK;                   // 512
    float s = 0.f, q = 0.f;
    const float* p = src + (size_t)chunk * rows * G4 + col;
    for (int r = 0; r < rows; ++r) {
        float v = p[(size_t)r * G4];
        s += v; q += v * v;
    }
    const size_t a = blockIdx.z;
    part[((a * 2 + 0) * NCHUNK + chunk) * (size_t)G4 + col] = s;
    part[((a * 2 + 1) * NCHUNK + chunk) * (size_t)G4 + col] = q;
}

// ---------------------------------------------------------------------------
// Kernel 3: finalize gate BN -> per-column affine (scale, shift).
// bn(v) = v*sc + t,  sc = g*rsqrt(var+eps),  t = beta - mean*sc  (+bias for ih)
// st layout: [s_ih(1024) t_ih(1024) s_hh(1024) t_hh(1024)]
// ---------------------------------------------------------------------------
__global__ __launch_bounds__(256)
void gate_stats_final(const float* __restrict__ part,
                      const float* __restrict__ gih, const float* __restrict__ bih,
                      const float* __restrict__ ghh, const float* __restrict__ bhh,
                      const float* __restrict__ bias, float* __restrict__ st)
{
    const int col = blockIdx.x * 256 + threadIdx.x;     // 0..1023
    const size_t a = blockIdx.y;
    float s = 0.f, q = 0.f;
    for (int ch = 0; ch < NCHUNK; ++ch) {               // fixed order: deterministic
        s += part[((a * 2 + 0) * NCHUNK + ch) * (size_t)G4 + col];
        q += part[((a * 2 + 1) * NCHUNK + ch) * (size_t)G4 + col];
    }
    const float mean = s * (1.f / BROWS);
    const float var  = q * (1.f / BROWS) - mean * mean; // biased variance
    const float g  = a ? ghh[col] : gih[col];
    const float be = a ? bhh[col] : bih[col];
    const float sc = g * rsqrtf(var + EPSV);
    float t = be - mean * sc;
    if (a == 0) t += bias[col];                          // fold LSTM bias into ih shift
    st[a * 2048 + col]        = sc;
    st[a * 2048 + 1024 + col] = t;
}

// ---------------------------------------------------------------------------
// Kernel 4: gates f,i,g -> pre-BN c_next written into d_out[0 .. B*H).
// ---------------------------------------------------------------------------
__global__ __launch_bounds__(256)
void cell_update(const float* __restrict__ ih, const float* __restrict__ hh,
                 const float* __restrict__ c_in, const float* __restrict__ st,
                 float* __restrict__ out)
{
    const size_t idx = (size_t)blockIdx.x * 256 + threadIdx.x;  // over B*H
    const int hcol = (int)(idx & (HID - 1));
    const size_t b = idx >> 8;
    const size_t base = b * G4 + hcol;
    const float* s_ih = st;
    const float* t_ih = st + 1024;
    const float* s_hh = st + 2048;
    const float* t_hh = st + 3072;

    float fg = ih[base +   0] * s_ih[hcol +   0] + t_ih[hcol +   0]
             + hh[base +   0] * s_hh[hcol +   0] + t_hh[hcol +   0];
    float ig = ih[base + 256] * s_ih[hcol + 256] + t_ih[hcol + 256]
             + hh[base + 256] * s_hh[hcol + 256] + t_hh[hcol + 256];
    float gg = ih[base + 768] * s_ih[hcol + 768] + t_ih[hcol + 768]
             + hh[base + 768] * s_hh[hcol + 768] + t_hh[hcol + 768];

    float f = 1.f / (1.f + __expf(-fg));
    float i = 1.f / (1.f + __expf(-ig));
    float g = tanhf(gg);
    out[idx] = f * c_in[idx] + i * g;
}

// ---------------------------------------------------------------------------
// Kernel 5/6: c_next BN stats (partial + finalize).  cpart: [sum|sq][chunk][256]
// ---------------------------------------------------------------------------
__global__ __launch_bounds__(256)
void c_stats_partial(const float* __restrict__ cpre, float* __restrict__ cpart)
{
    const int col = threadIdx.x;                        // 0..255
    const int chunk = blockIdx.x;
    const int rows = BROWS / NCHUNK;
    float s = 0.f, q = 0.f;
    const float* p = cpre + (size_t)chunk * rows * HID + col;
    for (int r = 0; r < rows; ++r) {
        float v = p[(size_t)r * HID];
        s += v; q += v * v;
    }
    cpart[(size_t)chunk * HID + col]                         = s;
    cpart[(size_t)NCHUNK * HID + (size_t)chunk * HID + col]  = q;
}

__global__ __launch_bounds__(256)
void c_stats_final(const float* __restrict__ cpart,
                   const float* __restrict__ gc, const float* __restrict__ bc,
                   float* __restrict__ st_c)
{
    const int col = threadIdx.x;
    float s = 0.f, q = 0.f;
    for (int ch = 0; ch < NCHUNK; ++ch) {
        s += cpart[(size_t)ch * HID + col];
        q += cpart[(size_t)NCHUNK * HID + (size_t)ch * HID + col];
    }
    const float mean = s * (1.f / BROWS);
    const float var  = q * (1.f / BROWS) - mean * mean;
    const float sc = gc[col] * rsqrtf(var + EPSV);
    st_c[col]       = sc;
    st_c[256 + col] = bc[col] - mean * sc;
}

// ---------------------------------------------------------------------------
// Kernel 7: normalize c_next in place; recompute o-gate from ih/hh; h_next.
// ---------------------------------------------------------------------------
__global__ __launch_bounds__(256)
void output_stage(const float* __restrict__ ih, const float* __restrict__ hh,
                  const float* __restrict__ st, const float* __restrict__ st_c,
                  float* __restrict__ out)
{
    const size_t idx = (size_t)blockIdx.x * 256 + threadIdx.x;
    const int hcol = (int)(idx & (HID - 1));
    const size_t b = idx >> 8;

    float cn = out[idx] * st_c[hcol] + st_c[256 + hcol];
    out[idx] = cn;                                      // normalized c_next

    const float* s_ih = st;
    const float* t_ih = st + 1024;
    const float* s_hh = st + 2048;
    const float* t_hh = st + 3072;
    const size_t base = b * G4 + 512 + hcol;            // o-gate columns
    float og = ih[base] * s_ih[512 + hcol] + t_ih[512 + hcol]
             + hh[base] * s_hh[512 + hcol] + t_hh[512 + hcol];
    float o = 1.f / (1.f + __expf(-og));
    out[(size_t)BROWS * HID + idx] = o * tanhf(cn);     // h_next
}

// ---------------------------------------------------------------------------
extern "C" void kernel_launch(void* const* d_in, const int* in_sizes, int n_in,
                              void* d_out, int out_size, void* d_ws, size_t ws_size,
                              hipStream_t stream)
{
    const float* x    = (const float*)d_in[0];
    const float* c    = (const float*)d_in[1];
    const float* h    = (const float*)d_in[2];
    const float* w_ih = (const float*)d_in[3];
    const float* w_hh = (const float*)d_in[4];
    const float* bias = (const float*)d_in[5];
    const float* g_ih = (const float*)d_in[6];
    const float* b_ih = (const float*)d_in[7];
    const float* g_hh = (const float*)d_in[8];
    const float* b_hh = (const float*)d_in[9];
    const float* g_c  = (const float*)d_in[10];
    const float* b_c  = (const float*)d_in[11];
    float* out = (float*)d_out;

    // workspace carve-up (floats)
    float* ws    = (float*)d_ws;
    float* ih    = ws;                                  // B*4H
    float* hh    = ih + (size_t)BROWS * G4;             // B*4H
    float* part  = hh + (size_t)BROWS * G4;             // 2*2*NCHUNK*1024
    float* st    = part + (size_t)2 * 2 * NCHUNK * G4;  // 4096
    float* cpart = st + 4096;                           // 2*NCHUNK*256
    float* st_c  = cpart + (size_t)2 * NCHUNK * HID;    // 512

    dim3 gGemm(BROWS / 128, G4 / 64, 2);
    gemm_gates<<<gGemm, 256, 0, stream>>>(x, h, w_ih, w_hh, ih, hh);

    dim3 gStat(G4 / 256, NCHUNK, 2);
    gate_stats_partial<<<gStat, 256, 0, stream>>>(ih, hh, part);

    dim3 gFin(G4 / 256, 2, 1);
    gate_stats_final<<<gFin, 256, 0, stream>>>(part, g_ih, b_ih, g_hh, b_hh, bias, st);

    const int nBH = BROWS * HID / 256;
    cell_update<<<nBH, 256, 0, stream>>>(ih, hh, c, st, out);

    c_stats_partial<<<NCHUNK, 256, 0, stream>>>(out, cpart);
    c_stats_final<<<1, 256, 0, stream>>>(cpart, g_c, b_c, st_c);

    output_stage<<<nBH, 256, 0, stream>>>(ih, hh, st, st_c, out);
}